// GCN_32375463477528
// MI455X (gfx1250) — compile-verified
//
#include <hip/hip_runtime.h>
#include <cmath>

// Problem dims (fixed by reference)
#define B_   16
#define O_   128
#define F_   2048
#define D_   128
#define OUTD (D_ + F_)   // 2176

typedef __attribute__((ext_vector_type(2))) float v2f;
typedef __attribute__((ext_vector_type(8))) float v8f;

// V_WMMA_F32_16X16X4_F32 : D(16x16 f32) = A(16x4 f32) * B(4x16 f32) + C
__device__ __forceinline__ v8f wmma4(v2f a, v2f b, v8f c) {
  return __builtin_amdgcn_wmma_f32_16x16x4_f32(
      false, a, false, b, (short)0, c, false, false);
}

// x0[b,j,:] = name_emb[j,:]  (broadcast over batch), float4 copies
__global__ void k_init_x(const float4* __restrict__ name_emb,
                         float4* __restrict__ x) {
  int t = blockIdx.x * blockDim.x + threadIdx.x;   // over B*O*D/4
  x[t] = name_emb[t & (O_ * D_ / 4 - 1)];
}

// R[b,j,d] = sum_i rel_table[obj_rels[b,i,j], d]
// one wave per (b,j); 32 lanes x float4 = full 512B row, coalesced.
__global__ __launch_bounds__(256) void k_R(const int* __restrict__ rels,
                                           const float4* __restrict__ rt4,
                                           float4* __restrict__ R4) {
  int wave = (blockIdx.x * blockDim.x + threadIdx.x) >> 5;  // b*O + j
  int lane = threadIdx.x & 31;                              // d/4
  int b = wave >> 7;
  int j = wave & (O_ - 1);
  const int* rp = rels + (size_t)b * O_ * O_ + j;           // stride O_ over i
  float4 s = make_float4(0.f, 0.f, 0.f, 0.f);
  for (int i = 0; i < O_; ++i) {
    int r = rp[(size_t)i * O_];                             // wave-uniform
    float4 v = rt4[(size_t)r * (D_ / 4) + lane];
    s.x += v.x; s.y += v.y; s.z += v.z; s.w += v.w;
  }
  R4[(size_t)wave * (D_ / 4) + lane] = s;
}

// dinv[b,k] = 1 / max(1, sum_j (rels[b,k,j] > 0))
__global__ void k_denom(const int* __restrict__ rels,
                        float* __restrict__ dinv) {
  int bk = blockIdx.x * blockDim.x + threadIdx.x;  // b*O + k
  const int* rp = rels + (size_t)bk * O_;
  int c = 0;
  for (int j = 0; j < O_; ++j) c += (rp[j] > 0) ? 1 : 0;
  dinv[bk] = 1.0f / (float)(c ? c : 1);
}

// One layer:
//   M[j,e]  = ((x[j,:] .* R[b,j,:]) @ Wb^T)[e] + O*bb[e]    (phase 1 -> LDS)
//   xdst[k] = tanh(x@Wa^T + ba + (A@M)*dinv + x)            (phase 2)
// One block per batch; 8 waves each own one 16-row tile.
__global__ __launch_bounds__(256) void k_layer(
    const float* __restrict__ xsrc, float* __restrict__ xdst,
    const float* __restrict__ R,    const float* __restrict__ dinv,
    const int*   __restrict__ rels,
    const float* __restrict__ Wa, const float* __restrict__ ba,
    const float* __restrict__ Wb, const float* __restrict__ bb) {
  __shared__ float Ms[O_ * D_];       // M[j][e], 64 KB (<< 320 KB/WGP)

  int b    = blockIdx.x;
  int tid  = threadIdx.x;
  int wave = tid >> 5;                // 0..7 (wave32)
  int lane = tid & 31;
  int half = lane >> 4;               // selects K pair in A/B frags
  int l16  = lane & 15;

  // ---------------- Phase 1: M into LDS ----------------
  {
    int jt   = wave;                  // j-tile
    int jrow = jt * 16 + l16;         // A-frag row (j) for this lane
    const float* xr = xsrc + ((size_t)b * O_ + jrow) * D_;
    const float* Rr = R    + ((size_t)b * O_ + jrow) * D_;

    // Preload A-side fragments (x .* R) once; reused across all 8 e-tiles.
    v2f av[32];
#pragma unroll
    for (int ks = 0; ks < 32; ++ks) {
      int k = ks * 4 + half * 2;
      float2 xv = *(const float2*)(xr + k);
      float2 rv = *(const float2*)(Rr + k);
      av[ks].x = xv.x * rv.x;
      av[ks].y = xv.y * rv.y;
    }

    for (int et = 0; et < 8; ++et) {  // e-tile
      int n = et * 16 + l16;          // B/C/D column for this lane
      const float* wbr = Wb + (size_t)n * D_;   // Wb[n, :] (B = Wb^T)
      v8f c = {};
#pragma unroll
      for (int ks = 0; ks < 32; ++ks) {         // K = 128 in steps of 4
        int k = ks * 4 + half * 2;
        float2 wv = *(const float2*)(wbr + k);
        v2f w; w.x = wv.x; w.y = wv.y;
        c = wmma4(av[ks], w, c);
      }
      float bias = (float)O_ * bb[n];           // sum_i of bb
#pragma unroll
      for (int r = 0; r < 8; ++r) {
        int m = jt * 16 + half * 8 + r;         // C/D layout: VGPR r, half
        Ms[m * D_ + n] = c[r] + bias;
      }
    }
  }
  __syncthreads();

  // ---------------- Phase 2: x update ----------------
  {
    int kt   = wave;                  // output row tile
    int krow = kt * 16 + l16;
    const float* xr = xsrc + ((size_t)b * O_ + krow) * D_;
    const int*   ar = rels + ((size_t)b * O_ + krow) * O_;  // A[krow, j]

    // Preload x-row A-fragments once (reused across 8 e-tiles).
    v2f a1v[32];
#pragma unroll
    for (int ks = 0; ks < 32; ++ks) {
      int k = ks * 4 + half * 2;
      float2 xv = *(const float2*)(xr + k);
      a1v[ks].x = xv.x; a1v[ks].y = xv.y;
    }
    // Pack adjacency-row fragment bits: 2 VGPRs instead of 64.
    unsigned m0 = 0u, m1 = 0u;
#pragma unroll
    for (int ks = 0; ks < 32; ++ks) {
      int k = ks * 4 + half * 2;
      int2 rv = *(const int2*)(ar + k);
      m0 |= (rv.x > 0 ? 1u : 0u) << ks;
      m1 |= (rv.y > 0 ? 1u : 0u) << ks;
    }
    // Per-output-row 1/denom, invariant over e-tiles.
    float dlv[8];
#pragma unroll
    for (int r = 0; r < 8; ++r)
      dlv[r] = dinv[b * O_ + kt * 16 + half * 8 + r];

    for (int et = 0; et < 8; ++et) {
      int n = et * 16 + l16;
      const float* war = Wa + (size_t)n * D_;
      v8f c1 = {};                    // x @ Wa^T tile
      v8f c2 = {};                    // A @ M tile
#pragma unroll
      for (int ks = 0; ks < 32; ++ks) {
        int k = ks * 4 + half * 2;
        float2 wv = *(const float2*)(war + k);
        v2f w1; w1.x = wv.x; w1.y = wv.y;
        c1 = wmma4(a1v[ks], w1, c1);
        v2f a2; a2.x = ((m0 >> ks) & 1u) ? 1.0f : 0.0f;
                a2.y = ((m1 >> ks) & 1u) ? 1.0f : 0.0f;
        v2f m2; m2.x = Ms[k * D_ + n];
                m2.y = Ms[(k + 1) * D_ + n];
        c2 = wmma4(a2, m2, c2);
      }
      float bn = ba[n];
#pragma unroll
      for (int r = 0; r < 8; ++r) {
        int m    = kt * 16 + half * 8 + r;
        float xv = xsrc[((size_t)b * O_ + m) * D_ + n];
        float v  = c1[r] + bn + c2[r] * dlv[r] + xv;
        xdst[((size_t)b * O_ + m) * D_ + n] = tanhf(v);
      }
    }
  }
}

// out[b,o,:] = concat(x[b,o,:], obj_feats[b,o,:])  -- the bandwidth term.
// float4 path: row strides 8704 / 512 / 8192 bytes, all 16B-aligned.
__global__ void k_out(const float4* __restrict__ x4,
                      const float4* __restrict__ f4,
                      float4* __restrict__ out4) {
  int row = blockIdx.x;               // b*O + o
  const float4* xr = x4  + (size_t)row * (D_ / 4);
  const float4* fr = f4  + (size_t)row * (F_ / 4);
  float4*     orow = out4 + (size_t)row * (OUTD / 4);
  for (int c = threadIdx.x; c < OUTD / 4; c += blockDim.x)
    orow[c] = (c < D_ / 4) ? xr[c] : fr[c - D_ / 4];
}

extern "C" void kernel_launch(void* const* d_in, const int* in_sizes, int n_in,
                              void* d_out, int out_size, void* d_ws, size_t ws_size,
                              hipStream_t stream) {
  (void)in_sizes; (void)n_in; (void)out_size; (void)ws_size;
  const float* obj_feats = (const float*)d_in[0];
  const int*   obj_rels  = (const int*)  d_in[1];  // int32 (JAX x64 off)
  const float* name_emb  = (const float*)d_in[2];
  const float* rel_table = (const float*)d_in[3];
  const float* W0 = (const float*)d_in[4];  const float* b0 = (const float*)d_in[5];
  const float* W1 = (const float*)d_in[6];  const float* b1 = (const float*)d_in[7];
  const float* W2 = (const float*)d_in[8];  const float* b2 = (const float*)d_in[9];
  const float* W3 = (const float*)d_in[10]; const float* b3 = (const float*)d_in[11];
  float* out = (float*)d_out;

  // Workspace layout (~3.02 MB total)
  float* xa = (float*)d_ws;                       // [B,O,D]
  float* xb = xa + (size_t)B_ * O_ * D_;          // [B,O,D]
  float* Rb = xb + (size_t)B_ * O_ * D_;          // [B,O,D]
  float* dv = Rb + (size_t)B_ * O_ * D_;          // [B,O]

  k_init_x<<<(B_ * O_ * D_ / 4) / 256, 256, 0, stream>>>(
      (const float4*)name_emb, (float4*)xa);
  k_R<<<(B_ * O_) / 8, 256, 0, stream>>>(          // 8 waves/block
      obj_rels, (const float4*)rel_table, (float4*)Rb);
  k_denom<<<B_, O_, 0, stream>>>(obj_rels, dv);
  // layer 1: (Wa,ba,Wb,bb) = (W0,b0,W1,b1)
  k_layer<<<B_, 256, 0, stream>>>(xa, xb, Rb, dv, obj_rels, W0, b0, W1, b1);
  // layer 2: (W2,b2,W3,b3)
  k_layer<<<B_, 256, 0, stream>>>(xb, xa, Rb, dv, obj_rels, W2, b2, W3, b3);
  k_out<<<B_ * O_, 256, 0, stream>>>(
      (const float4*)xa, (const float4*)obj_feats, (float4*)out);
}